// VertexNormals_57037165691511
// MI455X (gfx1250) — compile-verified
//
#include <hip/hip_runtime.h>
#include <cstdint>
#include <cstddef>

// Problem constants (match reference)
#define B_DIM 16
#define V_DIM 131072
#define F_DIM 262144
#define EPS_N 1e-12f

#define THREADS 256
#define TILE_FACES 256                         // one face per thread per tile
#define TILE_BYTES (TILE_FACES * 24)           // 3 x int64 per face = 6144 B
#define TILES_PER_BATCH (F_DIM / TILE_FACES)   // 1024
#define TILES_PER_BLOCK 16
#define BLOCKS_X (TILES_PER_BATCH / TILES_PER_BLOCK) // 64

// ---- gfx1250 async-copy path (ASYNCcnt / global_load_async_to_lds_*) ----
#if defined(__gfx1250__) &&                                                  \
    __has_builtin(__builtin_amdgcn_global_load_async_to_lds_b128) &&         \
    __has_builtin(__builtin_amdgcn_global_load_async_to_lds_b64)  &&         \
    __has_builtin(__builtin_amdgcn_s_wait_asynccnt)
#define USE_ASYNC_LDS 1
#else
#define USE_ASYNC_LDS 0
#endif

#if USE_ASYNC_LDS
// Builtin signatures (per clang diagnostics):
//   b64 : (v2i32 AS1*, v2i32 AS3*, imm int offset, imm int cpol)
//   b128: (v4i32 AS1*, v4i32 AS3*, imm int offset, imm int cpol)
typedef int v2i32 __attribute__((vector_size(8)));
typedef int v4i32 __attribute__((vector_size(16)));
typedef __attribute__((address_space(1))) v2i32 as1_v2i32;
typedef __attribute__((address_space(3))) v2i32 as3_v2i32;
typedef __attribute__((address_space(1))) v4i32 as1_v4i32;
typedef __attribute__((address_space(3))) v4i32 as3_v4i32;

__device__ __forceinline__ as1_v2i32* gbl_v2(const void* p) {
    return (as1_v2i32*)(uintptr_t)p;
}
__device__ __forceinline__ as1_v4i32* gbl_v4(const void* p) {
    return (as1_v4i32*)(uintptr_t)p;
}
// generic LDS pointer: low 32 bits are the LDS byte offset
__device__ __forceinline__ as3_v2i32* lds_v2(const void* p) {
    return (as3_v2i32*)(unsigned int)(uintptr_t)p;
}
__device__ __forceinline__ as3_v4i32* lds_v4(const void* p) {
    return (as3_v4i32*)(unsigned int)(uintptr_t)p;
}
#endif

// ---------------- zero accumulator ----------------
__global__ void __launch_bounds__(THREADS)
vn_zero_kernel(float4* __restrict__ p, int n4) {
    int i = blockIdx.x * blockDim.x + threadIdx.x;
    if (i < n4) p[i] = make_float4(0.f, 0.f, 0.f, 0.f);
}

// ---------------- per-face gather + cross + atomic scatter ----------------
__device__ __forceinline__ void process_face(const float* __restrict__ vb,
                                             float* __restrict__ ab,
                                             int i0, int i1, int i2) {
    const float* p0 = vb + (size_t)i0 * 3;
    const float* p1 = vb + (size_t)i1 * 3;
    const float* p2 = vb + (size_t)i2 * 3;
    float x0 = p0[0], y0 = p0[1], z0 = p0[2];
    float x1 = p1[0], y1 = p1[1], z1 = p1[2];
    float x2 = p2[0], y2 = p2[1], z2 = p2[2];

    float e0x = x1 - x0, e0y = y1 - y0, e0z = z1 - z0;
    float e1x = x2 - x1, e1y = y2 - y1, e1z = z2 - z1;
    float e2x = x0 - x2, e2y = y0 - y2, e2z = z0 - z2;

    float nx = (e0y * e1z - e0z * e1y) + (e1y * e2z - e1z * e2y) + (e2y * e0z - e2z * e0y);
    float ny = (e0z * e1x - e0x * e1z) + (e1z * e2x - e1x * e2z) + (e2z * e0x - e2x * e0z);
    float nz = (e0x * e1y - e0y * e1x) + (e1x * e2y - e1y * e2x) + (e2x * e0y - e2y * e0x);

    float* a0 = ab + (size_t)i0 * 3;
    float* a1 = ab + (size_t)i1 * 3;
    float* a2 = ab + (size_t)i2 * 3;
    unsafeAtomicAdd(a0 + 0, nx); unsafeAtomicAdd(a0 + 1, ny); unsafeAtomicAdd(a0 + 2, nz);
    unsafeAtomicAdd(a1 + 0, nx); unsafeAtomicAdd(a1 + 1, ny); unsafeAtomicAdd(a1 + 2, nz);
    unsafeAtomicAdd(a2 + 0, nx); unsafeAtomicAdd(a2 + 1, ny); unsafeAtomicAdd(a2 + 2, nz);
}

__global__ void __launch_bounds__(THREADS)
vn_face_scatter_kernel(const float* __restrict__ verts,
                       const long long* __restrict__ faces,
                       float* __restrict__ acc) {
    const int b = blockIdx.y;
    const float* vb = verts + (size_t)b * V_DIM * 3;
    float* ab = acc + (size_t)b * V_DIM * 3;
    const unsigned char* fb =
        (const unsigned char*)(faces + (size_t)b * F_DIM * 3);
    const int tid   = threadIdx.x;
    const int tile0 = blockIdx.x * TILES_PER_BLOCK;

#if USE_ASYNC_LDS
    __shared__ unsigned char smem[2][TILE_BYTES];

    { // prologue: issue tile0 (2 uniform async ops per thread: 8B + 16B = 24B/face)
        const unsigned char* src = fb + (size_t)tile0 * TILE_BYTES + tid * 24;
        unsigned char* dst = &smem[0][tid * 24];
        __builtin_amdgcn_global_load_async_to_lds_b64 (gbl_v2(src),     lds_v2(dst),     0, 0);
        __builtin_amdgcn_global_load_async_to_lds_b128(gbl_v4(src + 8), lds_v4(dst + 8), 0, 0);
    }

    for (int t = 0; t < TILES_PER_BLOCK; ++t) {
        if (t + 1 < TILES_PER_BLOCK) {
            const unsigned char* src =
                fb + (size_t)(tile0 + t + 1) * TILE_BYTES + tid * 24;
            unsigned char* dst = &smem[(t + 1) & 1][tid * 24];
            __builtin_amdgcn_global_load_async_to_lds_b64 (gbl_v2(src),     lds_v2(dst),     0, 0);
            __builtin_amdgcn_global_load_async_to_lds_b128(gbl_v4(src + 8), lds_v4(dst + 8), 0, 0);
            __builtin_amdgcn_s_wait_asynccnt(2); // current tile's 2 ops retired (in-order)
        } else {
            __builtin_amdgcn_s_wait_asynccnt(0);
        }
        __syncthreads(); // everyone's copies for tile t visible

        const unsigned int* fi = (const unsigned int*)&smem[t & 1][tid * 24];
        int i0 = (int)fi[0];  // low dword of int64 index (V < 2^31)
        int i1 = (int)fi[2];
        int i2 = (int)fi[4];
        process_face(vb, ab, i0, i1, i2);

        __syncthreads(); // buffer safe to overwrite next iteration
    }
#else
    for (int t = 0; t < TILES_PER_BLOCK; ++t) {
        const long long* fp =
            (const long long*)(fb + (size_t)(tile0 + t) * TILE_BYTES + tid * 24);
        int i0 = (int)__builtin_nontemporal_load(fp + 0);
        int i1 = (int)__builtin_nontemporal_load(fp + 1);
        int i2 = (int)__builtin_nontemporal_load(fp + 2);
        process_face(vb, ab, i0, i1, i2);
    }
#endif
}

// ---------------- per-vertex normalize (in place) ----------------
__global__ void __launch_bounds__(THREADS)
vn_normalize_kernel(float* __restrict__ out) {
    size_t i = (size_t)blockIdx.x * blockDim.x + threadIdx.x;
    if (i >= (size_t)B_DIM * V_DIM) return;
    float* p = out + i * 3;
    float x = p[0], y = p[1], z = p[2];
    float n = sqrtf(x * x + y * y + z * z);
    float inv = 1.0f / fmaxf(n, EPS_N);
    p[0] = x * inv;
    p[1] = y * inv;
    p[2] = z * inv;
}

extern "C" void kernel_launch(void* const* d_in, const int* in_sizes, int n_in,
                              void* d_out, int out_size, void* d_ws, size_t ws_size,
                              hipStream_t stream) {
    const float*     verts = (const float*)d_in[0];     // (B, V, 3) f32
    const long long* faces = (const long long*)d_in[1]; // (B, F, 3) i64
    float*           out   = (float*)d_out;             // (B, V, 3) f32

    // 1) zero the accumulator (d_out doubles as accumulator; harness poisons it)
    const int n4 = (B_DIM * V_DIM * 3) / 4; // 25,165,824 B / 16
    vn_zero_kernel<<<(n4 + THREADS - 1) / THREADS, THREADS, 0, stream>>>(
        (float4*)out, n4);

    // 2) face normals + atomic scatter-add (L2-resident accumulator)
    dim3 grid(BLOCKS_X, B_DIM);
    vn_face_scatter_kernel<<<grid, THREADS, 0, stream>>>(verts, faces, out);

    // 3) normalize in place
    const int nv = B_DIM * V_DIM;
    vn_normalize_kernel<<<(nv + THREADS - 1) / THREADS, THREADS, 0, stream>>>(out);
}